// PmModel_15169824489608
// MI455X (gfx1250) — compile-verified
//
#include <hip/hip_runtime.h>

typedef float v2f __attribute__((ext_vector_type(2)));
typedef float v4f __attribute__((ext_vector_type(4)));
typedef float v8f __attribute__((ext_vector_type(8)));

#define FDIM 128
#define TDIM 4
#define HDIM 32
#define WDIM 32
#define NPIX (TDIM * HDIM * WDIM)     // 4096 tokens
#define WAVES_PER_BLOCK 8             // 256 threads = 8 wave32

// One wave per token: out[n] = W[lin(n)] (128x128) @ x[n] (128) + b[lin(n)]
// WMMA_F32_16X16X4_F32 with x replicated across the 16 B-columns; D column 0
// is the matvec result. Bias is preloaded into the C accumulator.
__global__ __launch_bounds__(256)
void PmModel_pointwise_linear(const float* __restrict__ x,
                              const float* __restrict__ Wm,
                              const float* __restrict__ bias,
                              int* __restrict__ inds_out,
                              float* __restrict__ out)
{
    __shared__ float xs[WAVES_PER_BLOCK][FDIM];

    const int lane = threadIdx.x & 31;
    const int wave = threadIdx.x >> 5;
    const int n    = blockIdx.x * WAVES_PER_BLOCK + wave;   // token id (wave-uniform)
    const int half = lane >> 4;                             // 0: lanes 0-15, 1: 16-31
    const int lrow = lane & 15;

    // ---- inds output: (t, h, w) per token ----
    const int ti  = n / (HDIM * WDIM);
    const int rem = n % (HDIM * WDIM);
    const int hi  = rem / WDIM;
    const int wi  = rem % WDIM;
    // raster id exactly as reference: t*hw + h*H + w  (== n when H == W)
    const int lin = ti * (HDIM * WDIM) + hi * HDIM + wi;
    if (lane < 3) {
        inds_out[n * 3 + lane] = (lane == 0) ? ti : (lane == 1) ? hi : wi;
    }

    // ---- stage x[n, :] into LDS (one float4 per lane) ----
    const float* xrow = x + (long)n * FDIM;
    *(v4f*)&xs[wave][lane * 4] = *(const v4f*)(xrow + lane * 4);
    __syncthreads();

    // ---- preload all B operands: B[k, j] = x[k0 + k] for every column j ----
    // B (4x16 f32) half-wave layout: v0 = rows {0|2}, v1 = rows {1|3}
    //   lane<16 : {x[k0+0], x[k0+1]}   lane>=16 : {x[k0+2], x[k0+3]}
    v2f xb[32];
    #pragma unroll
    for (int kb = 0; kb < 32; ++kb) {
        xb[kb] = *(const v2f*)&xs[wave][kb * 4 + half * 2];   // ds_load_b64 broadcast
    }

    const float* Wrow = Wm   + (long)lin * FDIM * FDIM;  // this token's 128x128 matrix
    const float* brow = bias + (long)lin * FDIM;
    float*       orow = out  + (long)n * FDIM;

    for (int i0 = 0; i0 < 8; ++i0) {                     // 8 row-tiles of 16
        // C preloaded with bias tile: VGPR r = b[i0*16 + half*8 + r] (all cols equal)
        v8f acc = *(const v8f*)(brow + i0 * 16 + half * 8);

        // prefetch next row tile of the streamed 256MB W
        if (i0 < 7) {
            __builtin_prefetch(Wrow + (long)((i0 + 1) * 16 + lrow) * FDIM + half * 2, 0, 1);
        }

        #pragma unroll
        for (int kb = 0; kb < 32; ++kb) {
            // A (16x4 f32): lane L(0-15) = W[i0*16+L, 4kb+{0,1}], lane L+16 = +{2,3}
            const v2f a = __builtin_nontemporal_load(
                (const v2f*)(Wrow + (long)(i0 * 16 + lrow) * FDIM + kb * 4 + half * 2));
            acc = __builtin_amdgcn_wmma_f32_16x16x4_f32(
                false, a, false, xb[kb], (short)0, acc, false, false);
        }

        // D column 0 = out rows [i0*16, i0*16+16): lane 0 has rows m=0..7,
        // lane 16 has rows m=8..15 (VGPR r -> M = half*8 + r)
        if (lrow == 0) {
            *(v8f*)(orow + i0 * 16 + half * 8) = acc;
        }
    }
}

extern "C" void kernel_launch(void* const* d_in, const int* in_sizes, int n_in,
                              void* d_out, int out_size, void* d_ws, size_t ws_size,
                              hipStream_t stream) {
    const float* x = (const float*)d_in[0];   // [4,32,32,128]
    const float* W = (const float*)d_in[1];   // [4096,128,128]
    const float* b = (const float*)d_in[2];   // [4096,128]
    int*   inds_out = (int*)d_out;                    // [4096,3] int32 (raw bits)
    float* out      = (float*)d_out + NPIX * 3;       // [4096,128] f32
    PmModel_pointwise_linear<<<NPIX / WAVES_PER_BLOCK, 256, 0, stream>>>(
        x, W, b, inds_out, out);
}